// GaussianMixtureSpatialModel_12412455486063
// MI455X (gfx1250) — compile-verified
//
#include <hip/hip_runtime.h>
#include <hip/hip_bf16.h>

typedef __attribute__((ext_vector_type(2))) float v2f;
typedef __attribute__((ext_vector_type(8))) float v8f;

#define LOG2PI_F 1.8378770664093453f
#define SENTINEL (-3.0e38f)

static constexpr int T = 2048;
static constexpr int TILE = 16;

// One wave per (n, t-tile).  V_WMMA_F32_16X16X4_F32 computes, per 16x16 tile,
//   g_spatial[t,s] = -0.5*c1*sqdist(x_t,x_s)
// via the rank-4 factorization:
//   A row t = [c1*x0, c1*x1, -0.5*c1*|x|^2, 1]   (16x4)
//   B col s = [y0,    y1,    1,  -0.5*c1*|y|^2]  (4x16)
// loglik[t] = cst + LSE_{s<t}(g_spatial + t_s/b) - LSE_{s<t}(t_s/b),  b = softplus(cd)
__global__ __launch_bounds__(32)
void gmsm_hawkes_kernel(const float* __restrict__ times,   // [N,T]
                        const float* __restrict__ locs,    // [N,T,2]
                        const float* __restrict__ mask,    // [N,T]
                        const float* __restrict__ mu0p,    // [1]
                        const float* __restrict__ logstd0p,// [1]
                        const float* __restrict__ coeffp,  // [1]
                        const float* __restrict__ slogstdp,// [1]
                        float* __restrict__ out)           // [N,T]
{
    const int ttile = blockIdx.x;
    const int n     = blockIdx.y;
    const int lane  = threadIdx.x;        // 0..31 (wave32)
    const int half  = lane >> 4;
    const int l16   = lane & 15;

    const float sstd    = *slogstdp;
    const float c1      = __expf(-2.0f * sstd);
    const float cst     = -(2.0f * sstd + LOG2PI_F);             // -0.5*D*(2s+log2pi), D=2
    const float cd      = *coeffp;
    const float beta    = (cd > 20.0f) ? cd : log1pf(__expf(cd)); // softplus
    const float invbeta = 1.0f / beta;

    const float* tn = times + (size_t)n * T;
    const float* xn = locs  + (size_t)n * T * 2;

    const int t0 = ttile * TILE;

    // ---- A fragment (16x4 f32): lanes 0-15 (M=lane): V0=K0,V1=K1;
    //      lanes 16-31 (M=lane-16): V0=K2,V1=K3.
    v2f afrag;
    {
        const float x0 = xn[(t0 + l16) * 2 + 0];
        const float x1 = xn[(t0 + l16) * 2 + 1];
        if (half == 0) { afrag.x = c1 * x0;  afrag.y = c1 * x1; }
        else           { afrag.x = -0.5f * c1 * (x0 * x0 + x1 * x1); afrag.y = 1.0f; }
    }

    // Online-LSE state: g per row r (row t = t0 + r + 8*half, cols lane-distributed),
    // h shared across rows for full tiles (row-independent value).
    float mg[8], ag[8];
#pragma unroll
    for (int r = 0; r < 8; ++r) { mg[r] = SENTINEL; ag[r] = 0.0f; }
    float mh = SENTINEL, ah = 0.0f;

    // ================= full tiles: branchless =================
    for (int stile = 0; stile < ttile; ++stile) {
        const int s = stile * TILE + l16;
        const float y0 = xn[s * 2 + 0];
        const float y1 = xn[s * 2 + 1];
        const float ts = tn[s];

        v2f bfrag;
        if (half == 0) { bfrag.x = y0; bfrag.y = y1; }
        else           { bfrag.x = 1.0f; bfrag.y = -0.5f * c1 * (y0 * y0 + y1 * y1); }

        v8f cacc = {};
        v8f d = __builtin_amdgcn_wmma_f32_16x16x4_f32(
            false, afrag, false, bfrag, (short)0, cacc, false, false);

        const float vh = ts * invbeta;
        // shared h update (once per lane per tile)
        {
            const float mn = fmaxf(mh, vh);
            ah = ah * __expf(mh - mn) + __expf(vh - mn);
            mh = mn;
        }
#pragma unroll
        for (int r = 0; r < 8; ++r) {
            const float v  = d[r] + vh;
            const float mn = fmaxf(mg[r], v);
            ag[r] = ag[r] * __expf(mg[r] - mn) + __expf(v - mn);
            mg[r] = mn;
        }
    }

    // ================= diagonal tile: sentinel-masked, no exec branches =================
    float mhd[8], ahd[8];
    {
        const int s = t0 + l16;
        const float y0 = xn[s * 2 + 0];
        const float y1 = xn[s * 2 + 1];
        const float ts = tn[s];

        v2f bfrag;
        if (half == 0) { bfrag.x = y0; bfrag.y = y1; }
        else           { bfrag.x = 1.0f; bfrag.y = -0.5f * c1 * (y0 * y0 + y1 * y1); }

        v8f cacc = {};
        v8f d = __builtin_amdgcn_wmma_f32_16x16x4_f32(
            false, afrag, false, bfrag, (short)0, cacc, false, false);

        const float vh = ts * invbeta;
#pragma unroll
        for (int r = 0; r < 8; ++r) {
            // valid iff s < trow  <=>  l16 < r + 8*half
            const bool valid = (l16 < r + 8 * half);
            const float v   = valid ? (d[r] + vh) : SENTINEL;
            const float vhm = valid ? vh          : SENTINEL;
            // g: merge into running per-row state (invalid contributes (SENTINEL,1),
            // which is annihilated by any finite-max partial in the merges below)
            float mn = fmaxf(mg[r], v);
            ag[r] = ag[r] * __expf(mg[r] - mn) + __expf(v - mn);
            mg[r] = mn;
            // h (diagonal part, per row)
            mhd[r] = vhm;
            ahd[r] = 1.0f;
        }
    }

    // ---- per-row h = merge(shared full-tile h, diagonal h)
    float mh2[8], ah2[8];
#pragma unroll
    for (int r = 0; r < 8; ++r) {
        const float mn = fmaxf(mh, mhd[r]);
        ah2[r] = ah * __expf(mh - mn) + ahd[r] * __expf(mhd[r] - mn);
        mh2[r] = mn;
    }

    // ---- cross-lane LSE merge over 16 columns (xor offsets < 16 stay in each half-wave)
#pragma unroll
    for (int r = 0; r < 8; ++r) {
#pragma unroll
        for (int off = 1; off < 16; off <<= 1) {
            float mo = __shfl_xor(mg[r], off, 32);
            float ao = __shfl_xor(ag[r], off, 32);
            float mn = fmaxf(mg[r], mo);
            ag[r] = ag[r] * __expf(mg[r] - mn) + ao * __expf(mo - mn);
            mg[r] = mn;
            mo = __shfl_xor(mh2[r], off, 32);
            ao = __shfl_xor(ah2[r], off, 32);
            mn = fmaxf(mh2[r], mo);
            ah2[r] = ah2[r] * __expf(mh2[r] - mn) + ao * __expf(mo - mn);
            mh2[r] = mn;
        }
    }

    // ---- lane 0 of each half writes its 8 rows
    if (l16 == 0) {
#pragma unroll
        for (int r = 0; r < 8; ++r) {
            const int trow = t0 + r + 8 * half;
            float o;
            if (trow == 0) {
                const float mu0 = *mu0p, ls0 = *logstd0p;
                const float inv = __expf(-ls0);
                const float u0 = (xn[0] - mu0) * inv;
                const float u1 = (xn[1] - mu0) * inv;
                o = -0.5f * (u0 * u0 + 2.0f * ls0 + LOG2PI_F)
                    -0.5f * (u1 * u1 + 2.0f * ls0 + LOG2PI_F);
            } else {
                const float ll = cst + (mg[r] + __logf(ag[r])) - (mh2[r] + __logf(ah2[r]));
                o = ll * mask[(size_t)n * T + trow];
            }
            out[(size_t)n * T + trow] = o;
        }
    }
}

extern "C" void kernel_launch(void* const* d_in, const int* in_sizes, int n_in,
                              void* d_out, int out_size, void* d_ws, size_t ws_size,
                              hipStream_t stream) {
    (void)in_sizes; (void)n_in; (void)d_ws; (void)ws_size; (void)out_size;
    const float* times   = (const float*)d_in[0];
    const float* locs    = (const float*)d_in[1];
    const float* mask    = (const float*)d_in[2];
    const float* mu0     = (const float*)d_in[3];
    const float* logstd0 = (const float*)d_in[4];
    const float* coeff   = (const float*)d_in[5];
    const float* slogstd = (const float*)d_in[6];
    float* out = (float*)d_out;

    dim3 grid(T / TILE, 16);   // (t-tiles, N)
    dim3 block(32);            // one wave32
    gmsm_hawkes_kernel<<<grid, block, 0, stream>>>(times, locs, mask, mu0, logstd0,
                                                   coeff, slogstd, out);
}